// HamiltonianAttention_57260503990359
// MI455X (gfx1250) — compile-verified
//
#include <hip/hip_runtime.h>
#include <math.h>

#define BATCH   8
#define NSEQ    2048
#define DIMX    1024
#define HDIM    512
#define EPS_F   0.03f
#define SCL_F   (1.0f/32.0f)   // 1/sqrt(1024)

typedef float v2f __attribute__((ext_vector_type(2)));
typedef float v8f __attribute__((ext_vector_type(8)));

__device__ __forceinline__ v8f wmma_f32(v2f a, v2f b, v8f c) {
    // V_WMMA_F32_16X16X4_F32 : D = A(16x4) * B(4x16) + C(16x16)
    return __builtin_amdgcn_wmma_f32_16x16x4_f32(
        /*neg_a=*/false, a, /*neg_b=*/false, b,
        /*c_mod=*/(short)0, c, /*reuse_a=*/false, /*reuse_b=*/false);
}

// A-fragment (or B=M^T fragment) from row-major matrix, leading dim ld.
// lanes 0-15: row=lm, k={k0,k0+1}; lanes 16-31: row=lm, k={k0+2,k0+3}
__device__ __forceinline__ v2f fragA_rm(const float* p, int ld, int k0, int lm, int lh) {
    const float* q = p + (size_t)lm * ld + k0 + 2 * lh;
    v2f r; r.x = q[0]; r.y = q[1]; return r;
}

__device__ __forceinline__ v8f v8zero() {
    v8f z;
    #pragma unroll
    for (int j = 0; j < 8; ++j) z[j] = 0.0f;
    return z;
}

// ---------------------------------------------------------------------------
// Kernel 1: flash-attention forward.  Block = 256 thr (8 waves), 16 queries/WG.
// Wave w computes the 16-key S tile for keys kb+16w (no duplicated S GEMM) and
// owns dims [w*128, w*128+128) of the P@V accumulator.
// ---------------------------------------------------------------------------
__global__ __launch_bounds__(256)
void ham_fwd_kernel(const float* __restrict__ Q, const float* __restrict__ Km,
                    const float* __restrict__ Vm, float* __restrict__ attn_out,
                    float* __restrict__ mbuf, float* __restrict__ lbuf) {
    __shared__ float Ps[16 * 128];
    __shared__ float mrun[16], lrun[16], fac[16];

    const int b = blockIdx.y, q0 = blockIdx.x * 16;
    const int tid = threadIdx.x, w = tid >> 5, lane = tid & 31;
    const int lm = lane & 15, lh = lane >> 4;

    const float* Qb = Q  + ((size_t)(b * NSEQ + q0)) * DIMX;
    const float* Kb = Km + (size_t)b * NSEQ * DIMX;
    const float* Vb = Vm + (size_t)b * NSEQ * DIMX;

    if (tid < 16) { mrun[tid] = -INFINITY; lrun[tid] = 0.0f; }
    v8f O[8];
    #pragma unroll
    for (int t = 0; t < 8; ++t) O[t] = v8zero();
    __syncthreads();

    for (int kb = 0; kb < NSEQ; kb += 128) {
        const int kk = kb + 16 * w;
        const float* Krow = Kb + (size_t)kk * DIMX;

        // ---- S tile (16q x 16k), K-dim = 1024 -> 256 WMMAs ----
        v8f s = v8zero();
        for (int k0 = 0; k0 < DIMX; k0 += 4) {
            v2f a  = fragA_rm(Qb,   DIMX, k0, lm, lh);
            v2f bb = fragA_rm(Krow, DIMX, k0, lm, lh);   // B = K^T
            s = wmma_f32(a, bb, s);
        }

        __syncthreads();   // prior consumers of Ps are done
        #pragma unroll
        for (int j = 0; j < 8; ++j)
            Ps[(j + 8 * lh) * 128 + 16 * w + lm] = s[j] * SCL_F;
        __syncthreads();

        // ---- online softmax: half-wave lh owns row r = 2w+lh; lane lm owns
        //      8 columns; tree-reduce max & sum across the 16-lane group.
        {
            const int r = 2 * w + lh;
            float* Prow = &Ps[r * 128 + lm * 8];
            float mb = -INFINITY;
            #pragma unroll
            for (int c = 0; c < 8; ++c) mb = fmaxf(mb, Prow[c]);
            #pragma unroll
            for (int off = 1; off < 16; off <<= 1)
                mb = fmaxf(mb, __shfl_xor(mb, off, 16));
            const float mo = mrun[r];
            const float mn = fmaxf(mo, mb);
            const float f  = __expf(mo - mn);
            float ssum = 0.0f;
            #pragma unroll
            for (int c = 0; c < 8; ++c) {
                float p = __expf(Prow[c] - mn);
                Prow[c] = p;
                ssum += p;
            }
            #pragma unroll
            for (int off = 1; off < 16; off <<= 1)
                ssum += __shfl_xor(ssum, off, 16);
            if (lm == 0) {
                lrun[r] = lrun[r] * f + ssum;
                mrun[r] = mn;
                fac[r]  = f;
            }
        }
        __syncthreads();

        // ---- rescale accumulators ----
        float fr[8];
        #pragma unroll
        for (int j = 0; j < 8; ++j) fr[j] = fac[j + 8 * lh];
        #pragma unroll
        for (int t = 0; t < 8; ++t)
            #pragma unroll
            for (int j = 0; j < 8; ++j) O[t][j] *= fr[j];

        // ---- O(16 x own 128 dims) += P(16x128) @ V(128 x dims)
        //      k0-outer: one shared A-frag (P from LDS) feeds 8 WMMAs.
        const int dcol = w * 128;
        for (int k0 = 0; k0 < 128; k0 += 4) {
            v2f a = fragA_rm(Ps, 128, k0, lm, lh);
            const float* Vp = Vb + (size_t)(kb + k0 + 2 * lh) * DIMX + dcol + lm;
            #pragma unroll
            for (int t = 0; t < 8; ++t) {
                v2f bb; bb.x = Vp[t * 16]; bb.y = Vp[t * 16 + DIMX];
                O[t] = wmma_f32(a, bb, O[t]);
            }
        }
    }

    // ---- finalize: O /= l, store attn_out + stats
    float linv[8];
    #pragma unroll
    for (int j = 0; j < 8; ++j) linv[j] = 1.0f / lrun[j + 8 * lh];
    const int dcol = w * 128;
    #pragma unroll
    for (int t = 0; t < 8; ++t)
        #pragma unroll
        for (int j = 0; j < 8; ++j) {
            const int row = j + 8 * lh;
            attn_out[((size_t)(b * NSEQ + q0 + row)) * DIMX + dcol + t * 16 + lm] =
                O[t][j] * linv[j];
        }
    if (tid < 16) {
        mbuf[b * NSEQ + q0 + tid] = mrun[tid];
        lbuf[b * NSEQ + q0 + tid] = lrun[tid];
    }
}

// ---------------------------------------------------------------------------
// Kernel 2: g = W1 @ (W2 ⊙ silu'(attn_out@W1 + b1))  per row.
// Block = 256 thr, 16 rows/WG.  Phase1: Z=X@W1 (wave w -> 64 cols).
// Phase2: G = U@W1^T (wave w -> 128 dims).  Both k0-outer for A-frag reuse.
// ---------------------------------------------------------------------------
__global__ __launch_bounds__(256)
void ham_grad_mlp_kernel(const float* __restrict__ X, const float* __restrict__ W1,
                         const float* __restrict__ b1, const float* __restrict__ W2,
                         float* __restrict__ G) {
    __shared__ float Us[16 * HDIM];   // 32 KB

    const int tid = threadIdx.x, w = tid >> 5, lane = tid & 31;
    const int lm = lane & 15, lh = lane >> 4;
    const size_t r0 = (size_t)blockIdx.x * 16;
    const float* Xb = X + r0 * DIMX;

    // Phase 1: Z tile columns [w*64, w*64+64)
    {
        v8f z[4];
        #pragma unroll
        for (int t = 0; t < 4; ++t) z[t] = v8zero();
        for (int k0 = 0; k0 < DIMX; k0 += 4) {
            v2f a = fragA_rm(Xb, DIMX, k0, lm, lh);
            const float* Wr = W1 + (size_t)(k0 + 2 * lh) * HDIM + w * 64 + lm;
            #pragma unroll
            for (int t = 0; t < 4; ++t) {
                v2f bb; bb.x = Wr[t * 16]; bb.y = Wr[t * 16 + HDIM];
                z[t] = wmma_f32(a, bb, z[t]);
            }
        }
        #pragma unroll
        for (int t = 0; t < 4; ++t) {
            const int ncol = w * 64 + t * 16;
            const float w2v = W2[ncol + lm];
            const float b1v = b1[ncol + lm];
            #pragma unroll
            for (int j = 0; j < 8; ++j) {
                const float zv  = z[t][j] + b1v;
                const float sig = 1.0f / (1.0f + __expf(-zv));
                const float dsi = sig * (1.0f + zv * (1.0f - sig));   // silu'(z)
                Us[(j + 8 * lh) * HDIM + ncol + lm] = w2v * dsi;
            }
        }
    }
    __syncthreads();

    // Phase 2: G dims [w*128, w*128+128)
    {
        v8f g[8];
        #pragma unroll
        for (int t = 0; t < 8; ++t) g[t] = v8zero();
        for (int k0 = 0; k0 < HDIM; k0 += 4) {
            v2f a = fragA_rm(Us, HDIM, k0, lm, lh);
            // B = W1^T : B[k][n] = W1[(dcol+n)][k]
            const float* Wp = W1 + (size_t)(w * 128 + lm) * HDIM + k0 + 2 * lh;
            #pragma unroll
            for (int t = 0; t < 8; ++t) {
                v2f bb; bb.x = Wp[(size_t)t * 16 * HDIM]; bb.y = Wp[(size_t)t * 16 * HDIM + 1];
                g[t] = wmma_f32(a, bb, g[t]);
            }
        }
        #pragma unroll
        for (int t = 0; t < 8; ++t)
            #pragma unroll
            for (int j = 0; j < 8; ++j)
                G[(r0 + j + 8 * lh) * DIMX + w * 128 + t * 16 + lm] = g[t][j];
    }
}

// ---------------------------------------------------------------------------
// Kernel 3: flash backward for dQ + final output.
// rowdot[n] = g[n]·attn_out[n];  per key block: recompute A from stored m/l,
// dA = G·V^T fused in the same K-loop, dS = A*(dA-rowdot)*SCALE staged in LDS,
// dQ += dS @ K.  out = V + attn_out - EPS*dQ.
// ---------------------------------------------------------------------------
__global__ __launch_bounds__(256)
void ham_bwd_kernel(const float* __restrict__ Q, const float* __restrict__ Km,
                    const float* __restrict__ Vm, const float* __restrict__ G,
                    const float* __restrict__ attn_out,
                    const float* __restrict__ mbuf, const float* __restrict__ lbuf,
                    float* __restrict__ out) {
    __shared__ float dSs[16 * 128];
    __shared__ float rdpart[256];
    __shared__ float rowdot[16], mrow[16], lrow[16];

    const int b = blockIdx.y, q0 = blockIdx.x * 16;
    const int tid = threadIdx.x, w = tid >> 5, lane = tid & 31;
    const int lm = lane & 15, lh = lane >> 4;

    const float* Qb = Q  + ((size_t)(b * NSEQ + q0)) * DIMX;
    const float* Gb = G  + ((size_t)(b * NSEQ + q0)) * DIMX;
    const float* Kb = Km + (size_t)b * NSEQ * DIMX;
    const float* Vb = Vm + (size_t)b * NSEQ * DIMX;

    // rowdot[r] = dot(g_row, attn_out_row): 16 threads per row, 64 elems each
    {
        const int r = tid >> 4, cc = (tid & 15) * 64;
        const size_t base = ((size_t)(b * NSEQ + q0 + r)) * DIMX + cc;
        float p = 0.0f;
        for (int c = 0; c < 64; ++c) p += G[base + c] * attn_out[base + c];
        rdpart[tid] = p;
    }
    if (tid < 16) {
        mrow[tid] = mbuf[b * NSEQ + q0 + tid];
        lrow[tid] = lbuf[b * NSEQ + q0 + tid];
    }
    __syncthreads();
    if (tid < 16) {
        float s = 0.0f;
        for (int i = 0; i < 16; ++i) s += rdpart[tid * 16 + i];
        rowdot[tid] = s;
    }
    __syncthreads();

    float m_j[8], linv_j[8], rd_j[8];
    #pragma unroll
    for (int j = 0; j < 8; ++j) {
        const int row = j + 8 * lh;
        m_j[j]    = mrow[row];
        linv_j[j] = 1.0f / lrow[row];
        rd_j[j]   = rowdot[row];
    }

    v8f dQ[8];
    #pragma unroll
    for (int t = 0; t < 8; ++t) dQ[t] = v8zero();

    for (int kb = 0; kb < NSEQ; kb += 128) {
        const int kk = kb + 16 * w;
        const float* Krow = Kb + (size_t)kk * DIMX;
        const float* Vrow = Vb + (size_t)kk * DIMX;

        // S = Q K^T  and  dA = G V^T  fused over K-dim 1024
        v8f s = v8zero(), da = v8zero();
        for (int k0 = 0; k0 < DIMX; k0 += 4) {
            v2f aq = fragA_rm(Qb,   DIMX, k0, lm, lh);
            v2f bk = fragA_rm(Krow, DIMX, k0, lm, lh);
            s = wmma_f32(aq, bk, s);
            v2f ag = fragA_rm(Gb,   DIMX, k0, lm, lh);
            v2f bv = fragA_rm(Vrow, DIMX, k0, lm, lh);
            da = wmma_f32(ag, bv, da);
        }

        __syncthreads();   // prior dSs consumers done
        #pragma unroll
        for (int j = 0; j < 8; ++j) {
            const float P = __expf(s[j] * SCL_F - m_j[j]) * linv_j[j];
            dSs[(j + 8 * lh) * 128 + 16 * w + lm] = P * (da[j] - rd_j[j]) * SCL_F;
        }
        __syncthreads();

        // dQ(16 x own 128 dims) += dS(16x128) @ K(128 x dims)
        // k0-outer: one shared A-frag (dS from LDS) feeds 8 WMMAs.
        const int dcol = w * 128;
        for (int k0 = 0; k0 < 128; k0 += 4) {
            v2f a = fragA_rm(dSs, 128, k0, lm, lh);
            const float* Kp = Kb + (size_t)(kb + k0 + 2 * lh) * DIMX + dcol + lm;
            #pragma unroll
            for (int t = 0; t < 8; ++t) {
                v2f bb; bb.x = Kp[t * 16]; bb.y = Kp[t * 16 + DIMX];
                dQ[t] = wmma_f32(a, bb, dQ[t]);
            }
        }
    }

    // out = V + attn_out - EPS * dQ
    #pragma unroll
    for (int t = 0; t < 8; ++t)
        #pragma unroll
        for (int j = 0; j < 8; ++j) {
            const int row = j + 8 * lh;
            const size_t idx =
                ((size_t)(b * NSEQ + q0 + row)) * DIMX + w * 128 + t * 16 + lm;
            out[idx] = Vm[idx] + attn_out[idx] - EPS_F * dQ[t][j];
        }
}

// ---------------------------------------------------------------------------
extern "C" void kernel_launch(void* const* d_in, const int* in_sizes, int n_in,
                              void* d_out, int out_size, void* d_ws, size_t ws_size,
                              hipStream_t stream) {
    const float* Q  = (const float*)d_in[0];
    const float* Km = (const float*)d_in[1];
    const float* Vm = (const float*)d_in[2];
    const float* W1 = (const float*)d_in[3];
    const float* b1 = (const float*)d_in[4];
    const float* W2 = (const float*)d_in[5];
    // d_in[6] = b2 : constant offset, no effect on output

    const size_t nrow = (size_t)BATCH * NSEQ;
    float* attn = (float*)d_ws;                 // BATCH*NSEQ*DIMX
    float* G    = attn + nrow * DIMX;           // BATCH*NSEQ*DIMX
    float* mb   = G + nrow * DIMX;              // BATCH*NSEQ
    float* lb   = mb + nrow;                    // BATCH*NSEQ

    dim3 blk(256);
    dim3 grdA(NSEQ / 16, BATCH);
    dim3 grdM((unsigned)(nrow / 16));

    ham_fwd_kernel<<<grdA, blk, 0, stream>>>(Q, Km, Vm, attn, mb, lb);
    ham_grad_mlp_kernel<<<grdM, blk, 0, stream>>>(attn, W1, b1, W2, G);
    ham_bwd_kernel<<<grdA, blk, 0, stream>>>(Q, Km, Vm, G, attn, mb, lb,
                                             (float*)d_out);
}